// Encoder_3934190043972
// MI455X (gfx1250) — compile-verified
//
#include <hip/hip_runtime.h>
#include <hip/hip_bf16.h>

typedef __attribute__((ext_vector_type(2))) float v2f;
typedef __attribute__((ext_vector_type(8))) float v8f;

#define DFEAT 128  // feature width being aggregated (both layers aggregate 128 floats)

// ---------------- degree / normalization ----------------

__global__ void k_init_deg(float* deg, int n) {
    int i = blockIdx.x * blockDim.x + threadIdx.x;
    if (i < n) deg[i] = 1.0f;  // self-loop contributes 1 to in-degree
}

__global__ void k_count_deg(const int* __restrict__ dst, float* deg, int e) {
    int i = blockIdx.x * blockDim.x + threadIdx.x;
    if (i < e) atomicAdd(&deg[dst[i]], 1.0f);
}

__global__ void k_rsqrt(float* deg, int n) {
    int i = blockIdx.x * blockDim.x + threadIdx.x;
    if (i < n) {
        float d = deg[i];
        deg[i] = (d > 0.0f) ? rsqrtf(d) : 0.0f;  // becomes dinv in place
    }
}

// ---------------- weight transpose (once per call, 32K elements) ----------------

__global__ void k_transpose(const float* __restrict__ in, float* __restrict__ outT,
                            int R, int Ccols) {
    int idx = blockIdx.x * blockDim.x + threadIdx.x;
    if (idx >= R * Ccols) return;
    int r = idx / Ccols;
    int c = idx - r * Ccols;
    outT[c * R + r] = in[idx];
}

// ---------------- aggregation (gather + scatter-add), 128 features ----------------

// out[i, :] = dinv[i]^2 * feat[i, :]   (self-loop term; also zero-initializes accumulator)
__global__ void k_self_init(const float* __restrict__ feat, const float* __restrict__ dinv,
                            float* __restrict__ out, int n) {
    int idx = blockIdx.x * blockDim.x + threadIdx.x;   // over n * 32 float4 chunks
    if (idx >= n * (DFEAT / 4)) return;
    int node = idx >> 5;                               // 32 float4 per row
    float w = dinv[node];
    w = w * w;
    const float4* f4 = (const float4*)feat;
    float4 v = f4[idx];
    float4 r = make_float4(w * v.x, w * v.y, w * v.z, w * v.w);
    ((float4*)out)[idx] = r;
}

// per edge: out[dst, :] += dinv[src]*dinv[dst] * feat[src, :]   (one wave32 per edge)
__global__ void k_edge_agg(const int* __restrict__ src, const int* __restrict__ dst,
                           const float* __restrict__ dinv, const float* __restrict__ feat,
                           float* __restrict__ out, int e) {
    int wave = blockIdx.x * (blockDim.x / 32) + (threadIdx.x / 32);
    int lane = threadIdx.x & 31;
    if (wave >= e) return;
    int s = src[wave];
    int d = dst[wave];
    float w = dinv[s] * dinv[d];
    const float4* fx = (const float4*)(feat + (size_t)s * DFEAT);
    float4 v = fx[lane];                               // 32 lanes x 4 floats = 128
    float* o = out + (size_t)d * DFEAT + lane * 4;
    atomicAdd(o + 0, w * v.x);
    atomicAdd(o + 1, w * v.y);
    atomicAdd(o + 2, w * v.z);
    atomicAdd(o + 3, w * v.w);
}

// ---------------- fp32 WMMA GEMM: C[M,NC] = A[M,K] x B[K,NC] (+bias, relu) ----------------
// B is pre-transposed: BT[NC][K], so every operand load is a contiguous b64 with
// compile-time immediate offsets. One wave computes a 16x64 strip: 4 column tiles
// sharing each A fragment (4 accumulators).
// ISA layouts (wave32), per k-step of 4:
//   A 16x4 : lane m=lane&15; hi=lane>>4; VGPR j holds A[m][k + 2*hi + j]
//   B 4x16 : lane n=lane&15; VGPR j holds B[k + 2*hi + j][n]  (= BT[n][k + 2*hi + j])
//   C 16x16: VGPR v holds C[v + 8*hi][n]
template <int K, int NC>
__global__ __launch_bounds__(128) void gcn_gemm_wmma(
    const float* __restrict__ A, const float* __restrict__ BT,
    const float* __restrict__ bias, float* __restrict__ C,
    int M, int apply_relu) {
    constexpr int TN = 4;                              // column tiles per wave
    int cg = blockIdx.x;                               // 64-wide column group
    int tile_m = blockIdx.y * blockDim.y + threadIdx.y;
    if (tile_m * 16 >= M) return;                      // wave-uniform exit

    int lane = threadIdx.x;                            // 0..31
    int nl = lane & 15;
    int hi = lane >> 4;
    int row_base = tile_m * 16;

    int arow = row_base + nl;
    if (arow > M - 1) arow = M - 1;                    // safe clamp (stores guarded)
    const float* Ap = A + (size_t)arow * K + 2 * hi;
    const float* Bp = BT + (size_t)(cg * (TN * 16) + nl) * K + 2 * hi;

    v8f acc0 = {}, acc1 = {}, acc2 = {}, acc3 = {};
#pragma unroll
    for (int k = 0; k < K; k += 4) {
        v2f a  = *(const v2f*)(Ap + k);
        v2f b0 = *(const v2f*)(Bp + 0 * 16 * K + k);
        v2f b1 = *(const v2f*)(Bp + 1 * 16 * K + k);
        v2f b2 = *(const v2f*)(Bp + 2 * 16 * K + k);
        v2f b3 = *(const v2f*)(Bp + 3 * 16 * K + k);
        acc0 = __builtin_amdgcn_wmma_f32_16x16x4_f32(false, a, false, b0, (short)0, acc0, false, false);
        acc1 = __builtin_amdgcn_wmma_f32_16x16x4_f32(false, a, false, b1, (short)0, acc1, false, false);
        acc2 = __builtin_amdgcn_wmma_f32_16x16x4_f32(false, a, false, b2, (short)0, acc2, false, false);
        acc3 = __builtin_amdgcn_wmma_f32_16x16x4_f32(false, a, false, b3, (short)0, acc3, false, false);
    }

#pragma unroll
    for (int t = 0; t < TN; ++t) {
        v8f acc = (t == 0) ? acc0 : (t == 1) ? acc1 : (t == 2) ? acc2 : acc3;
        int col = cg * (TN * 16) + t * 16 + nl;
        float bs = bias ? bias[col] : 0.0f;
#pragma unroll
        for (int v = 0; v < 8; ++v) {
            int row = row_base + v + 8 * hi;
            if (row < M) {
                float val = acc[v] + bs;
                if (apply_relu) val = fmaxf(val, 0.0f);
                C[(size_t)row * NC + col] = val;
            }
        }
    }
}

// ---------------- final epilogue: out = relu(out + b2) ----------------

__global__ void k_bias_relu(float* __restrict__ out, const float* __restrict__ b, int total) {
    int i = blockIdx.x * blockDim.x + threadIdx.x;
    if (i < total) out[i] = fmaxf(out[i] + b[i & (DFEAT - 1)], 0.0f);
}

// ---------------- host orchestration ----------------

extern "C" void kernel_launch(void* const* d_in, const int* in_sizes, int n_in,
                              void* d_out, int out_size, void* d_ws, size_t ws_size,
                              hipStream_t stream) {
    const float* x  = (const float*)d_in[0];   // [N,128]
    const int*   ei = (const int*)d_in[1];     // [2,E]
    const float* W1 = (const float*)d_in[2];   // [128,256]
    const float* b1 = (const float*)d_in[3];   // [256]
    const float* W2 = (const float*)d_in[4];   // [256,128]
    const float* b2 = (const float*)d_in[5];   // [128]
    float* out = (float*)d_out;                // [N,128]

    int N = in_sizes[0] / 128;
    int E = in_sizes[1] / 2;
    const int* src = ei;
    const int* dst = ei + E;

    // scratch (floats): dinv[N] | bufA[N*128] | h1[N*256] | W1T[128*256] | W2T[256*128]
    float* ws   = (float*)d_ws;
    float* dinv = ws;
    float* bufA = ws + N;
    float* h1   = bufA + (size_t)N * DFEAT;
    float* W1T  = h1 + (size_t)N * 256;
    float* W2T  = W1T + 128 * 256;

    const int BT = 256;

    // normalization: deg (with self loop) -> dinv ; transpose weights
    k_init_deg<<<(N + BT - 1) / BT, BT, 0, stream>>>(dinv, N);
    k_count_deg<<<(E + BT - 1) / BT, BT, 0, stream>>>(dst, dinv, E);
    k_rsqrt<<<(N + BT - 1) / BT, BT, 0, stream>>>(dinv, N);
    k_transpose<<<(128 * 256 + BT - 1) / BT, BT, 0, stream>>>(W1, W1T, 128, 256);
    k_transpose<<<(256 * 128 + BT - 1) / BT, BT, 0, stream>>>(W2, W2T, 256, 128);

    // layer 1: aggregate X (128 feats) first, then GEMM (+b1, relu)
    int chunks = N * (DFEAT / 4);
    k_self_init<<<(chunks + BT - 1) / BT, BT, 0, stream>>>(x, dinv, bufA, N);
    k_edge_agg<<<(E + 7) / 8, BT, 0, stream>>>(src, dst, dinv, x, bufA, E);

    dim3 blk(32, 4);                                    // 4 waves per block
    int tm = (N + 15) / 16;
    dim3 grid1(256 / 64, (tm + 3) / 4);
    gcn_gemm_wmma<128, 256><<<grid1, blk, 0, stream>>>(bufA, W1T, b1, h1, N, 1);

    // layer 2: GEMM to 128 first (no bias), then aggregate, then +b2, relu
    dim3 grid2(128 / 64, (tm + 3) / 4);
    gcn_gemm_wmma<256, 128><<<grid2, blk, 0, stream>>>(h1, W2T, nullptr, bufA, N, 0);

    k_self_init<<<(chunks + BT - 1) / BT, BT, 0, stream>>>(bufA, dinv, out, N);
    k_edge_agg<<<(E + 7) / 8, BT, 0, stream>>>(src, dst, dinv, bufA, out, E);

    int total = N * DFEAT;
    k_bias_relu<<<(total + BT - 1) / BT, BT, 0, stream>>>(out, b2, total);
}